// Encoder_Datasets_Publications_GCN_82257213653409
// MI455X (gfx1250) — compile-verified
//
#include <hip/hip_runtime.h>
#include <hip/hip_bf16.h>
#include <cstddef>

// ---------------------------------------------------------------------------
// GCN encoder for MI455X (gfx1250, wave32).
//   h1 = Agg(x @ W1) + b1 ; h2 = Agg(h1 @ W2) + b2
//   mu = h2 @ Wmu + bmu   ; lv = h2 @ Wlv + blv
// GEMMs use V_WMMA_F32_16X16X4_F32 (full f32 precision, matches reference).
// Aggregation = self-loop-init + per-edge atomic scatter (memory bound).
// ---------------------------------------------------------------------------

typedef __attribute__((ext_vector_type(2))) float v2f;
typedef __attribute__((ext_vector_type(4))) float v4f;
typedef __attribute__((ext_vector_type(8))) float v8f;

// ---------------- degree / norm ----------------

__global__ void deg_init_kernel(float* __restrict__ deg, int n) {
    int i = blockIdx.x * blockDim.x + threadIdx.x;
    if (i < n) deg[i] = 1.0f;   // self-loop contributes 1 to every node degree
}

__global__ void deg_edges_kernel(const int* __restrict__ dst, float* __restrict__ deg, int e) {
    int i = blockIdx.x * blockDim.x + threadIdx.x;
    if (i < e) atomicAdd(&deg[dst[i]], 1.0f);
}

__global__ void deg_rsqrt_kernel(float* __restrict__ deg, int n) {
    int i = blockIdx.x * blockDim.x + threadIdx.x;
    if (i < n) deg[i] = rsqrtf(deg[i]);   // deg -> dinv in place
}

// ---------------- WMMA f32 GEMM: C[M x NCOL] = A[M x K] * B[K x NCOL] (+bias) ----
// One wave computes one 16x16 tile of C. blockDim = (32, NCOL/16).
// M must be a multiple of 16 (N_NODES = 50000 = 16 * 3125) so EXEC is all 1s.
//
// V_WMMA_F32_16X16X4_F32 operand layout (ISA 7.12.2, 32-bit A 16x4):
//   lane l: m = l&15, kb = (l>>4)*2 ; A frag = { A[m][k+kb], A[m][k+kb+1] }
//   B frag (4x16, row striped across lanes): { B[k+kb][n], B[k+kb+1][n] }, n = l&15
//   C/D:   VGPR v, lane l -> C[v + 8*(l>>4)][l&15]

template <int K, int NCOL, bool ADD_BIAS>
__global__ __launch_bounds__(32 * (NCOL / 16))
void gemm_wmma_kernel(const float* __restrict__ A, const float* __restrict__ B,
                      const float* __restrict__ bias, float* __restrict__ C, int M) {
    const int lane = threadIdx.x;            // 0..31
    const int wv   = threadIdx.y;            // column tile within block
    const int row0 = blockIdx.x * 16;
    const int col0 = wv * 16;
    if (row0 >= M) return;                   // uniform across wave (grid == M/16 anyway)

    const int mn = lane & 15;                // M index (A) / N index (B,C)
    const int kb = (lane >> 4) * 2;          // per-lane K sub-offset {0,2}

    const float* Arow = A + (size_t)(row0 + mn) * K + kb;           // 8B aligned
    const float* Bp   = B + (size_t)kb * NCOL + col0 + mn;

    v8f acc = {0.f, 0.f, 0.f, 0.f, 0.f, 0.f, 0.f, 0.f};

#pragma unroll 4
    for (int k = 0; k < K; k += 4) {
        v2f a = *(const v2f*)(Arow + k);                 // global_load_b64
        v2f b;
        b.x = Bp[k * NCOL];
        b.y = Bp[(k + 1) * NCOL];
        acc = __builtin_amdgcn_wmma_f32_16x16x4_f32(
            /*neg_a=*/false, a, /*neg_b=*/false, b,
            /*c_mod=*/(short)0, acc, /*reuse_a=*/false, /*reuse_b=*/false);
    }

    const int mhalf = (lane >> 4) * 8;
    float bv = ADD_BIAS ? bias[col0 + mn] : 0.0f;
#pragma unroll
    for (int v = 0; v < 8; ++v) {
        C[(size_t)(row0 + mhalf + v) * NCOL + col0 + mn] = acc[v] + bv;
    }
}

// ---------------- aggregation ----------------
// out[i][f] = t[i][f]*dinv[i]^2 + bias[f]   (self-loop term + layer bias)
template <int F>
__global__ void agg_init_kernel(const float* __restrict__ t, const float* __restrict__ dinv,
                                const float* __restrict__ bias, float* __restrict__ out, int n) {
    size_t idx = (size_t)blockIdx.x * blockDim.x + threadIdx.x;
    if (idx < (size_t)n * F) {
        int i = (int)(idx >> __builtin_ctz(F));
        int f = (int)(idx & (F - 1));
        float di = dinv[i];
        out[idx] = t[idx] * di * di + bias[f];
    }
}

// one wave per edge: gather transformed source row, atomic-scatter into dst row
template <int F>   // F = 128 -> float4/lane, F = 64 -> float2/lane
__global__ void scatter_edges_kernel(const int* __restrict__ src, const int* __restrict__ dst,
                                     const float* __restrict__ dinv, const float* __restrict__ t,
                                     float* __restrict__ out, int e) {
    constexpr int VEC = F / 32;
    const int lane = threadIdx.x & 31;
    const int wave = threadIdx.x >> 5;
    const int eid  = blockIdx.x * (blockDim.x >> 5) + wave;
    if (eid >= e) return;

    int s = src[eid];
    int d = dst[eid];
    float w = dinv[s] * dinv[d];

    const float* tp = t + (size_t)s * F + lane * VEC;
    float*       op = out + (size_t)d * F + lane * VEC;

    if constexpr (VEC == 4) {
        v4f v = *(const v4f*)tp;                          // global_load_b128
        atomicAdd(op + 0, v.x * w);
        atomicAdd(op + 1, v.y * w);
        atomicAdd(op + 2, v.z * w);
        atomicAdd(op + 3, v.w * w);
    } else {
        v2f v = *(const v2f*)tp;                          // global_load_b64
        atomicAdd(op + 0, v.x * w);
        atomicAdd(op + 1, v.y * w);
    }
}

// ---------------------------------------------------------------------------

extern "C" void kernel_launch(void* const* d_in, const int* in_sizes, int n_in,
                              void* d_out, int out_size, void* d_ws, size_t ws_size,
                              hipStream_t stream) {
    constexpr int IN_D = 256, HID1 = 128, HID2 = 64, OUT_D = 32;

    const float* x   = (const float*)d_in[0];
    const int*   ei  = (const int*)d_in[1];     // [2][E] : row0 = src, row1 = dst
    const float* W1  = (const float*)d_in[2];
    const float* b1  = (const float*)d_in[3];
    const float* W2  = (const float*)d_in[4];
    const float* b2  = (const float*)d_in[5];
    const float* Wmu = (const float*)d_in[6];
    const float* bmu = (const float*)d_in[7];
    const float* Wlv = (const float*)d_in[8];
    const float* blv = (const float*)d_in[9];

    const int N = in_sizes[0] / IN_D;           // 50000 (multiple of 16)
    const int E = in_sizes[1] / 2;              // 800000
    const int* src = ei;
    const int* dst = ei + E;

    // workspace layout (f32)
    float* wsf  = (float*)d_ws;
    float* dinv = wsf;                            // N
    float* t1   = dinv + N;                       // N*HID1
    float* h1   = t1 + (size_t)N * HID1;          // N*HID1
    float* t2   = h1 + (size_t)N * HID1;          // N*HID2
    float* h2   = t2 + (size_t)N * HID2;          // N*HID2

    float* mu = (float*)d_out;                    // N*OUT_D
    float* lv = mu + (size_t)N * OUT_D;           // N*OUT_D

    // 1) degrees -> dinv
    deg_init_kernel<<<(N + 255) / 256, 256, 0, stream>>>(dinv, N);
    deg_edges_kernel<<<(E + 255) / 256, 256, 0, stream>>>(dst, dinv, E);
    deg_rsqrt_kernel<<<(N + 255) / 256, 256, 0, stream>>>(dinv, N);

    const int rowTiles = N / 16;

    // 2) layer 1: t1 = x @ W1 ; h1 = Agg(t1) + b1
    gemm_wmma_kernel<IN_D, HID1, false>
        <<<rowTiles, dim3(32, HID1 / 16), 0, stream>>>(x, W1, nullptr, t1, N);
    agg_init_kernel<HID1>
        <<<(int)(((size_t)N * HID1 + 255) / 256), 256, 0, stream>>>(t1, dinv, b1, h1, N);
    scatter_edges_kernel<HID1>
        <<<(E + 7) / 8, 256, 0, stream>>>(src, dst, dinv, t1, h1, E);

    // 3) layer 2: t2 = h1 @ W2 ; h2 = Agg(t2) + b2
    gemm_wmma_kernel<HID1, HID2, false>
        <<<rowTiles, dim3(32, HID2 / 16), 0, stream>>>(h1, W2, nullptr, t2, N);
    agg_init_kernel<HID2>
        <<<(int)(((size_t)N * HID2 + 255) / 256), 256, 0, stream>>>(t2, dinv, b2, h2, N);
    scatter_edges_kernel<HID2>
        <<<(E + 7) / 8, 256, 0, stream>>>(src, dst, dinv, t2, h2, E);

    // 4) heads: mu = h2 @ Wmu + bmu ; lv = h2 @ Wlv + blv
    gemm_wmma_kernel<HID2, OUT_D, true>
        <<<rowTiles, dim3(32, OUT_D / 16), 0, stream>>>(h2, Wmu, bmu, mu, N);
    gemm_wmma_kernel<HID2, OUT_D, true>
        <<<rowTiles, dim3(32, OUT_D / 16), 0, stream>>>(h2, Wlv, blv, lv, N);
}